// LossFunction_33827162423915
// MI455X (gfx1250) — compile-verified
//
#include <hip/hip_runtime.h>
#include <hip/hip_bf16.h>
#include <stdint.h>

// ---------------------------------------------------------------------------
// YOLO-style loss, MI455X (gfx1250). Memory-bound streaming reduction:
//  - async global->LDS (CDNA5 GLOBAL_LOAD_ASYNC_TO_LDS_B128, non-temporal,
//    pipelined with ASYNCcnt split waits)
//  - wave32 shuffle reductions
//  - deterministic two-pass reduction (no float atomics)
// Roofline: 2 x 136 MB read once -> ~11.7 us floor at 23.3 TB/s; compute
// (~0.2 GFLOP + 34M transcendentals) is negligible. No matmul structure, so
// WMMA is inapplicable; async DMA + wave32 reductions are the gfx1250 path.
// ---------------------------------------------------------------------------

#define N_IMG   4096
#define S_GRID  19
#define CH      23                       // B*5 + C = 2*5 + 13
#define TOTAL_CELLS (N_IMG * S_GRID * S_GRID)   // 1,478,656
#define CHUNK   128                      // cells per pipeline stage
#define BLOCK   128                      // threads per block (4 waves, wave32)
#define GRID    1024                     // partial-sum blocks
#define NCHUNK  (TOTAL_CELLS / CHUNK)    // 11,552 (exact)
#define F4_PER_CHUNK (CHUNK * CH / 4)    // 736 float4 transfers per tensor-chunk
#define LOADS_PER_THREAD 6               // ceil(736/128); padded to uniform count

__device__ __forceinline__ void async_copy_b128(uint32_t lds_byte_off,
                                                const void* gaddr) {
  // VDST = LDS byte address (workgroup-relative), VADDR = 64-bit global addr.
  // ASYNCcnt increments once per instruction per wave (12 per chunk below).
  // Data is read exactly once -> non-temporal hint keeps L2 unpolluted.
  asm volatile("global_load_async_to_lds_b128 %0, %1, off th:TH_LOAD_NT"
               :: "v"(lds_byte_off),
                  "v"((unsigned long long)(uintptr_t)gaddr)
               : "memory");
}
__device__ __forceinline__ void wait_async_le12() {
  asm volatile("s_wait_asynccnt 0xc" ::: "memory");
}
__device__ __forceinline__ void wait_async_le0() {
  asm volatile("s_wait_asynccnt 0x0" ::: "memory");
}

__device__ __forceinline__ float sigm(float x) {
  return 1.0f / (1.0f + __expf(-x));
}

__device__ __forceinline__ void issue_chunk(const float4* p4, const float4* g4,
                                            int chunk, uint32_t lp0, uint32_t lg0,
                                            uint32_t ldump, int tid) {
  const float4* gp = p4 + (size_t)chunk * F4_PER_CHUNK;
  const float4* gg = g4 + (size_t)chunk * F4_PER_CHUNK;
#pragma unroll
  for (int j = 0; j < LOADS_PER_THREAD; ++j) {
    const int idx = tid + j * BLOCK;
    const int ok  = idx < F4_PER_CHUNK;
    const int gi  = ok ? idx : 0;                 // valid addr even when padded
    async_copy_b128(ok ? lp0 + (uint32_t)idx * 16u : ldump,        gp + gi);
    async_copy_b128(ok ? lg0 + (uint32_t)idx * 16u : ldump + 16u,  gg + gi);
  }
}

__global__ void __launch_bounds__(BLOCK)
yolo_loss_partial(const float* __restrict__ pred,
                  const float* __restrict__ gt,
                  float* __restrict__ part) {
  __shared__ __align__(16) float shp[2][CHUNK * CH];   // 2 x 11776 B
  __shared__ __align__(16) float shg[2][CHUNK * CH];   // 2 x 11776 B
  __shared__ __align__(16) float shdump[8];            // async pad target
  __shared__ float red[BLOCK / 32][6];

  const int tid = threadIdx.x;
  const int bid = blockIdx.x;
  const float4* p4 = reinterpret_cast<const float4*>(pred);
  const float4* g4 = reinterpret_cast<const float4*>(gt);
  const uint32_t lp[2] = { (uint32_t)(uintptr_t)&shp[0][0],
                           (uint32_t)(uintptr_t)&shp[1][0] };
  const uint32_t lg[2] = { (uint32_t)(uintptr_t)&shg[0][0],
                           (uint32_t)(uintptr_t)&shg[1][0] };
  const uint32_t ld = (uint32_t)(uintptr_t)&shdump[0];

  float sxy = 0.f, swh = 0.f, sconf = 0.f, scls = 0.f, snoobj = 0.f, cobj = 0.f;

  // prologue: start first chunk's DMA
  issue_chunk(p4, g4, bid, lp[0], lg[0], ld, tid);

  int buf = 0;
  for (int c = bid; c < NCHUNK; c += GRID, buf ^= 1) {
    const int nc = c + GRID;
    if (nc < NCHUNK) {
      issue_chunk(p4, g4, nc, lp[buf ^ 1], lg[buf ^ 1], ld, tid);
      wait_async_le12();     // 12 newer outstanding -> chunk c fully in LDS
    } else {
      wait_async_le0();
    }
    __syncthreads();

    const float* p = &shp[buf][tid * CH];   // stride 23 dwords: conflict-free
    const float* g = &shg[buf][tid * CH];

    float pb[10];
#pragma unroll
    for (int k = 0; k < 10; ++k) pb[k] = sigm(p[k]);

    float iou[2];
#pragma unroll
    for (int b = 0; b < 2; ++b) {
      const float px = pb[5*b+0], py = pb[5*b+1], pw = pb[5*b+2], ph = pb[5*b+3];
      const float gx = g[5*b+0],  gy = g[5*b+1],  gw = g[5*b+2],  gh = g[5*b+3];
      // faithful to reference (including its x/y-mixed mask and wh mixing)
      const float b1x0 = px - 0.5f*pw, b1y0 = py - 0.5f*ph;
      const float b1x1 = px + 0.5f*gw, b1y1 = py + 0.5f*gh;
      const float b2x0 = gx - 0.5f*pw, b2y0 = gy - 0.5f*ph;
      const float b2x1 = gx + 0.5f*gw, b2y1 = gy + 0.5f*gh;
      const bool m = (b1x0 - b2y0 >= 0.f) || (b2x0 - b1y0 >= 0.f) ||
                     (b2x1 - b1y1 >= 0.f) || (b1x1 - b2y1 >= 0.f);
      const float a1 = (b1x1 - b1x0) * (b1y1 - b1y0);
      const float a2 = (b2x1 - b2x0) * (b2y1 - b2y0);
      const float ix0 = fmaxf(b1x0, b2x0), iy0 = fmaxf(b1y0, b2y0);
      const float ix1 = fminf(b1x1, b2x1), iy1 = fminf(b1y1, b2y1);
      const float inter = (ix1 - ix0) * (iy1 - iy0);
      const float v = inter / (a1 + a2 - inter + 1e-6f);
      iou[b] = m ? 0.f : v;
    }
    const int   r    = (iou[1] > iou[0]) ? 1 : 0;  // argmax, first-max on ties
    const float riou = iou[r];
    const float gc   = g[4];
    if (gc > 0.f) {
      cobj += 1.f;
      const float dx = pb[5*r+0] - g[5*r+0], dy = pb[5*r+1] - g[5*r+1];
      sxy += dx*dx + dy*dy;
      const float dw = pb[5*r+2] - g[5*r+2], dh = pb[5*r+3] - g[5*r+3];
      swh += dw*dw + dh*dh;
      const float dc = pb[5*r+4] - riou;
      sconf += dc*dc;
      float mx = p[10];
#pragma unroll
      for (int j = 1; j < 13; ++j) mx = fmaxf(mx, p[10+j]);
      float e[13], es = 0.f;
#pragma unroll
      for (int j = 0; j < 13; ++j) { e[j] = __expf(p[10+j] - mx); es += e[j]; }
      const float inv = 1.f / es;
#pragma unroll
      for (int j = 0; j < 13; ++j) {
        const float d = e[j] * inv - g[10+j];
        scls += d * d;
      }
    } else if (gc == 0.f) {
      const float d0 = p[4] - g[4], d1 = p[9] - g[9];   // RAW preds
      snoobj += d0*d0 + d1*d1;
    }
    __syncthreads();   // buffer consumed before it may be re-filled
  }

  // wave32 reduction, then cross-wave via LDS (deterministic order)
#pragma unroll
  for (int off = 16; off > 0; off >>= 1) {
    sxy    += __shfl_down(sxy,    off, 32);
    swh    += __shfl_down(swh,    off, 32);
    sconf  += __shfl_down(sconf,  off, 32);
    scls   += __shfl_down(scls,   off, 32);
    snoobj += __shfl_down(snoobj, off, 32);
    cobj   += __shfl_down(cobj,   off, 32);
  }
  const int wave = tid >> 5, lane = tid & 31;
  if (lane == 0) {
    red[wave][0] = sxy;   red[wave][1] = swh;    red[wave][2] = sconf;
    red[wave][3] = scls;  red[wave][4] = snoobj; red[wave][5] = cobj;
  }
  __syncthreads();
  if (tid < 6) {
    part[bid * 6 + tid] =
        red[0][tid] + red[1][tid] + red[2][tid] + red[3][tid];
  }
}

__global__ void __launch_bounds__(256)
yolo_loss_finalize(const float* __restrict__ part, float* __restrict__ out) {
  __shared__ float red[8][6];
  const int tid = threadIdx.x;
  float a[6] = {0.f, 0.f, 0.f, 0.f, 0.f, 0.f};
  for (int i = tid; i < GRID; i += 256) {
#pragma unroll
    for (int k = 0; k < 6; ++k) a[k] += part[i * 6 + k];
  }
#pragma unroll
  for (int off = 16; off > 0; off >>= 1) {
#pragma unroll
    for (int k = 0; k < 6; ++k) a[k] += __shfl_down(a[k], off, 32);
  }
  const int wave = tid >> 5, lane = tid & 31;
  if (lane == 0) {
#pragma unroll
    for (int k = 0; k < 6; ++k) red[wave][k] = a[k];
  }
  __syncthreads();
  if (tid == 0) {
    float s[6];
#pragma unroll
    for (int k = 0; k < 6; ++k) {
      float t = 0.f;
      for (int w = 0; w < 8; ++w) t += red[w][k];
      s[k] = t;
    }
    const float n_obj   = s[5];
    const float n_noobj = (float)TOTAL_CELLS - n_obj;
    const float denom2  = fmaxf(2.0f * n_obj, 1.0f);
    const float xy_loss    = s[0] / denom2;
    const float wh_loss    = s[1] / denom2;
    const float conf_loss  = s[2] / fmaxf(n_obj, 1.0f);
    const float class_loss = s[3] / fmaxf(13.0f * n_obj, 1.0f);
    const float noobj_mse  = s[4] / fmaxf(2.0f * n_noobj, 1.0f);
    const float noobj_loss = noobj_mse * 0.5f;     // LAMBDA_NOOBJ applied twice,
    out[0] = 5.0f * (xy_loss + wh_loss) + conf_loss
           + 0.5f * noobj_loss + class_loss;       // faithful to reference
  }
}

extern "C" void kernel_launch(void* const* d_in, const int* in_sizes, int n_in,
                              void* d_out, int out_size, void* d_ws, size_t ws_size,
                              hipStream_t stream) {
  (void)in_sizes; (void)n_in; (void)out_size; (void)ws_size;
  const float* pred = (const float*)d_in[0];
  const float* gt   = (const float*)d_in[1];
  float* part = (float*)d_ws;                      // GRID*6 floats = 24 KB
  yolo_loss_partial<<<GRID, BLOCK, 0, stream>>>(pred, gt, part);
  yolo_loss_finalize<<<1, 256, 0, stream>>>(part, (float*)d_out);
}